// MaxPool2d_53901839564893
// MI455X (gfx1250) — compile-verified
//
#include <hip/hip_runtime.h>

typedef float v4f __attribute__((ext_vector_type(4)));
typedef float v2f __attribute__((ext_vector_type(2)));

// Input  x  : (32,64,256,256) f32, viewed as (NC=2048, 256, 256)
// Output out: (NC=2048, 128, 128)
//
// One work item = one (nc, oh, ow-pair). 1,048,576 threads (4096 blocks x 256),
// 16 iterations each, covering 2048*128*64 = 16,777,216 items exactly.
//
// Thread t (t < 2^20) owns fixed (ow2 = t&63, oh = (t>>6)&127) and channels
// nc = (t>>13) + 128*i, i = 0..15. So both pointers advance by a constant
// stride per iteration: +128*65536 floats (in), +128*16384 floats (out).
// Loop body is pure streaming: 2 prefetches, 2 NT b128 loads, 6 max, 1 NT
// b64 store, 2 pointer bumps.

__global__ __launch_bounds__(256) void maxpool2x2_f32_kernel(
    const float* __restrict__ x, float* __restrict__ out) {
  const int idx = (int)(blockIdx.x * 256u + threadIdx.x);  // 0 .. 2^20-1
  const int ow2 = idx & 63;          // pair of output columns
  const int oh  = (idx >> 6) & 127;  // output row
  const int nc0 = idx >> 13;         // starting channel, 0..127

  const float* r0 = x + (size_t)nc0 * 65536u + (size_t)(oh * 2) * 256u
                      + (size_t)(ow2 * 4);
  float* op = out + (size_t)nc0 * 16384u + (size_t)oh * 128u
                  + (size_t)(ow2 * 2);

  constexpr size_t IN_STEP  = (size_t)128 * 65536u;  // 32 MiB / 4
  constexpr size_t OUT_STEP = (size_t)128 * 16384u;

#pragma unroll 2
  for (int i = 0; i < 16; ++i) {
    // gfx1250 VMEM prefetch of next iteration's two cachelines (speculative:
    // past-the-end translation failures silently dropped, ISA 10.5).
    __builtin_prefetch(r0 + IN_STEP, 0, 0);
    __builtin_prefetch(r0 + IN_STEP + 256, 0, 0);

    // Streaming, zero-reuse data (512 MiB > 192 MiB L2): non-temporal.
    v4f a = __builtin_nontemporal_load((const v4f*)r0);          // row 2h
    v4f b = __builtin_nontemporal_load((const v4f*)(r0 + 256));  // row 2h+1

    v2f r;
    r.x = fmaxf(fmaxf(a.x, a.y), fmaxf(b.x, b.y));
    r.y = fmaxf(fmaxf(a.z, a.w), fmaxf(b.z, b.w));

    __builtin_nontemporal_store(r, (v2f*)op);

    r0 += IN_STEP;
    op += OUT_STEP;
  }
}

extern "C" void kernel_launch(void* const* d_in, const int* in_sizes, int n_in,
                              void* d_out, int out_size, void* d_ws, size_t ws_size,
                              hipStream_t stream) {
  (void)in_sizes; (void)n_in; (void)d_ws; (void)ws_size; (void)out_size;
  const float* x = (const float*)d_in[0];
  float* out = (float*)d_out;
  // 4096 blocks * 256 threads (8 wave32s/block), 16 iterations/thread, exact.
  maxpool2x2_f32_kernel<<<dim3(4096), dim3(256), 0, stream>>>(x, out);
}